// BCJRDecoder_61727269978271
// MI455X (gfx1250) — compile-verified
//
#include <hip/hip_runtime.h>
#include <hip/hip_bf16.h>

// BCJR MAP decoder for a 64-state rate-1/2 convolutional code.
// Mapping: one wave32 per batch element (grid = 256 blocks x 32 threads).
// Lane l owns trellis states 2l and 2l+1. Cross-state traffic uses wave
// shuffles; per-128-step segment tables (branch metrics bm, a-priori ap)
// live in LDS. Backward pass uses forward checkpoints every SEG steps and
// recomputes alphas per segment into LDS (no global workspace, ~39KB LDS).
// Branch-metric einsum ('btj,oj->tbo') is computed with
// V_WMMA_F32_16X16X4_F32 (16 timesteps per wmma, K=2 real + 2 zero-pad).

typedef __attribute__((ext_vector_type(2))) float v2f;
typedef __attribute__((ext_vector_type(8))) float v8f;

#define T_LEN 2048
#define N_LEN 4096
#define NSTATE 64
#define SEG 128
#define NSEG (T_LEN / SEG) // 16

// Output symbol index for (state s, input bit i), derived from
// GEN_POLY = ('1111001','1011011'):
//   o0 = i ^ parity(s & 0b111001), o1 = i ^ parity(s & 0b011011)
__device__ __forceinline__ int op_idx(int s, int i) {
    int o0 = i ^ (__popc(s & 57) & 1);
    int o1 = i ^ (__popc(s & 27) & 1);
    return (o0 << 1) | o1;
}

__device__ __forceinline__ float wave_sum(float v) {
#pragma unroll
    for (int off = 16; off >= 1; off >>= 1)
        v += __shfl_xor(v, off, 32);
    return v;
}

__global__ __launch_bounds__(32) void bcjr_kernel(const float* __restrict__ llr_ch,
                                                  const float* __restrict__ llr_a,
                                                  float* __restrict__ out) {
    const int b = blockIdx.x;
    const int lane = threadIdx.x;
    const float* ch = llr_ch + (size_t)b * N_LEN;   // 2 LLRs per trellis step
    const float* la = llr_a + (size_t)b * T_LEN;    // a-priori LLR per step
    float* outp = out + (size_t)b * T_LEN;

    __shared__ float s_bm[SEG * 4];        // branch metrics exp(0.5*(±l0±l1))
    __shared__ float s_ap0[SEG];           // exp(+0.5*la)
    __shared__ float s_ap1[SEG];           // exp(-0.5*la)
    __shared__ float s_alpha[SEG * NSTATE];// recomputed alphas (entering each t)
    __shared__ float s_cp[NSEG * NSTATE];  // forward checkpoints (segment entry)

    // ---- per-lane trellis constants (hoisted out of the time loops) ----
    // Forward: lane l produces new states j0=2l, j1=2l+1.
    //   input bit on transitions into j: bj = j>>5 (same for j0,j1)
    //   predecessors of j: p0=2*(j&31), p1=p0+1 -> pair owned by lane (j&31)
    const int bj = (lane >= 16) ? 1 : 0;
    const int src0 = (2 * lane) & 31;
    const int src1 = (2 * lane + 1) & 31;
    const int o00 = op_idx(2 * src0, bj);
    const int o01 = op_idx(2 * src0 + 1, bj);
    const int o10 = op_idx(2 * src1, bj);
    const int o11 = op_idx(2 * src1 + 1, bj);
    // Backward: states s0=2l, s1=2l+1; to(s,i) = (i<<5)|(s>>1) = l or l+32.
    const int oA = op_idx(2 * lane, 0);      // i=1 symbol is oA^3
    const int oB = op_idx(2 * lane + 1, 0);

    // WMMA B operand: B[0][n] = 0.5*sgn0(n), B[1][n] = 0.5*sgn1(n), rows 2,3 = 0.
    // B (4x16 f32, 2 VGPRs): v0 = rows K0|K2 (lanes 0-15 | 16-31), v1 = K1|K3.
    v2f bmat;
    bmat.x = (lane < 16) ? 0.5f * (1.0f - 2.0f * (float)((lane >> 1) & 1)) : 0.0f;
    bmat.y = (lane < 16) ? 0.5f * (1.0f - 2.0f * (float)(lane & 1)) : 0.0f;

    // ---- per-segment table builder: a-priori terms + WMMA branch metrics ----
    auto seg_tables = [&](int t0) {
        __syncthreads(); // previous consumers of s_bm/s_ap done (single-wave WG: cheap)
#pragma unroll
        for (int k = 0; k < SEG / 32; ++k) {
            int tt = k * 32 + lane;
            float e = expf(0.5f * la[t0 + tt]);
            s_ap0[tt] = e;
            s_ap1[tt] = 1.0f / e;
        }
#pragma unroll
        for (int g = 0; g < SEG / 16; ++g) {
            // A (16x4 f32): lanes 0-15 hold K=0,1 in v0,v1; lanes 16-31 hold K=2,3 (zero pad)
            v2f a;
            if (lane < 16) {
                int t = t0 + g * 16 + lane;
                a.x = ch[2 * t];
                a.y = ch[2 * t + 1];
            } else {
                a.x = 0.0f;
                a.y = 0.0f;
            }
            v8f c = {0.0f, 0.0f, 0.0f, 0.0f, 0.0f, 0.0f, 0.0f, 0.0f};
            // D = A x B : D[m][n] = 0.5*(l0[m]*sgn0(n) + l1[m]*sgn1(n)), n<4 valid
            c = __builtin_amdgcn_wmma_f32_16x16x4_f32(false, a, false, bmat,
                                                      (short)0, c, false, false);
            // D layout: VGPR r = row m=r (lanes 0-15) / m=8+r (lanes 16-31), N=lane&15.
            if ((lane < 4) || (lane >= 16 && lane < 20)) {
                int n = lane & 3;
                int mb = (lane >= 16) ? 8 : 0;
#pragma unroll
                for (int r = 0; r < 8; ++r)
                    s_bm[(g * 16 + mb + r) * 4 + n] = expf(c[r]);
            }
        }
        __syncthreads();
    };

    // ---- forward recursion step (normalized, matching reference) ----
    auto fwd_step = [&](float& ae, float& ao, int tt) {
        float pe0 = __shfl(ae, src0, 32);
        float po0 = __shfl(ao, src0, 32);
        float pe1 = __shfl(ae, src1, 32);
        float po1 = __shfl(ao, src1, 32);
        const float* bmt = s_bm + tt * 4;
        float ap = bj ? s_ap1[tt] : s_ap0[tt];
        float ne = ap * (bmt[o00] * pe0 + bmt[o01] * po0);
        float no = ap * (bmt[o10] * pe1 + bmt[o11] * po1);
        float inv = 1.0f / wave_sum(ne + no);
        ae = ne * inv;
        ao = no * inv;
    };

    // ================= Pass A: forward, record segment checkpoints ==========
    float ae = (lane == 0) ? 1.0f : 0.0f;
    float ao = 0.0f;
    for (int seg = 0; seg < NSEG; ++seg) {
        s_cp[seg * NSTATE + 2 * lane] = ae;       // entering alpha (same-lane reuse)
        s_cp[seg * NSTATE + 2 * lane + 1] = ao;
        seg_tables(seg * SEG);
        for (int tt = 0; tt < SEG; ++tt)
            fwd_step(ae, ao, tt);
    }

    // ================= Pass B: backward with per-segment alpha recompute ====
    float be = 1.0f / 64.0f;
    float bo = 1.0f / 64.0f;
    for (int seg = NSEG - 1; seg >= 0; --seg) {
        int t0 = seg * SEG;
        seg_tables(t0);
        // recompute alphas entering each step of this segment
        float fe = s_cp[seg * NSTATE + 2 * lane];
        float fo = s_cp[seg * NSTATE + 2 * lane + 1];
        for (int tt = 0; tt < SEG; ++tt) {
            s_alpha[tt * NSTATE + 2 * lane] = fe;
            s_alpha[tt * NSTATE + 2 * lane + 1] = fo;
            fwd_step(fe, fo, tt);
        }
        // backward sweep
        for (int tt = SEG - 1; tt >= 0; --tt) {
            // beta_next[l] and beta_next[l+32] (shared by both states in lane)
            float e = __shfl(be, lane >> 1, 32);
            float o = __shfl(bo, lane >> 1, 32);
            float bn0 = (lane & 1) ? o : e;
            float e2 = __shfl(be, 16 + (lane >> 1), 32);
            float o2 = __shfl(bo, 16 + (lane >> 1), 32);
            float bn1 = (lane & 1) ? o2 : e2;

            const float* bmt = s_bm + tt * 4;
            float ap0 = s_ap0[tt], ap1 = s_ap1[tt];
            float bg00 = ap0 * bmt[oA] * bn0;        // bg[s0][i=0]
            float bg01 = ap1 * bmt[oA ^ 3] * bn1;    // bg[s0][i=1]
            float bg10 = ap0 * bmt[oB] * bn0;        // bg[s1][i=0]
            float bg11 = ap1 * bmt[oB ^ 3] * bn1;    // bg[s1][i=1]

            float aet = s_alpha[tt * NSTATE + 2 * lane];
            float aot = s_alpha[tt * NSTATE + 2 * lane + 1];
            float j0 = wave_sum(aet * bg00 + aot * bg10);
            float j1 = wave_sum(aet * bg01 + aot * bg11);

            float nbe = bg00 + bg01;
            float nbo = bg10 + bg11;
            float inv = 1.0f / wave_sum(nbe + nbo);
            if (lane == 0)
                outp[t0 + tt] = logf(j0) - logf(j1);
            be = nbe * inv;
            bo = nbo * inv;
        }
    }
}

extern "C" void kernel_launch(void* const* d_in, const int* in_sizes, int n_in,
                              void* d_out, int out_size, void* d_ws, size_t ws_size,
                              hipStream_t stream) {
    const float* llr_ch = (const float*)d_in[0]; // (B, 4096) f32
    const float* llr_a = (const float*)d_in[1];  // (B, 2048) f32
    float* out = (float*)d_out;                  // (B, 2048) f32
    int B = in_sizes[0] / N_LEN; // 256
    bcjr_kernel<<<dim3(B), dim3(32), 0, stream>>>(llr_ch, llr_a, out);
}